// IouLoss_9088150798817
// MI455X (gfx1250) — compile-verified
//
#include <hip/hip_runtime.h>
#include <cstdint>

typedef __attribute__((ext_vector_type(2))) float v2f;
typedef __attribute__((ext_vector_type(8))) float v8f;

#define BLOCK       256
#define MAX_GT_TILE 1024
#define IOU_EPS     1e-6f

__global__ __launch_bounds__(BLOCK) void iou_loss_kernel(
    const float* __restrict__ proposals,
    const float* __restrict__ gt,
    float* __restrict__ out,
    int n_p, int n_g)
{
    __shared__ float4 s_gt[MAX_GT_TILE];     // 16 KB tile of gt boxes
    __shared__ float  s_ag[MAX_GT_TILE];     // precomputed gt areas
    __shared__ float  s_wsum[BLOCK / 32];    // per-wave partials (wave32)

    const int tid  = threadIdx.x;
    const int gidx = blockIdx.x * BLOCK + tid;

    // One proposal per thread, coalesced 16B load. Keep ALL threads live
    // (EXEC must be all-ones at the WMMA); invalid lanes contribute 0.
    const bool valid = (gidx < n_p);
    float4 p = make_float4(0.f, 0.f, 0.f, 0.f);
    if (valid) p = ((const float4*)proposals)[gidx];
    const float area_p_eps = (p.z - p.x) * (p.w - p.y) + IOU_EPS;

    float max_iou = 0.0f;

    for (int base = 0; base < n_g; base += MAX_GT_TILE) {
        int cnt = n_g - base;
        if (cnt > MAX_GT_TILE) cnt = MAX_GT_TILE;

        __syncthreads();  // previous tile fully consumed before overwrite

        // ---- CDNA5 async copy: global -> LDS, tracked by ASYNCcnt ----
        for (int j = tid; j < cnt; j += BLOCK) {
            unsigned lds_addr = (unsigned)(size_t)(&s_gt[j]);   // low 32b = LDS offset
            const float4* gaddr = ((const float4*)gt) + base + j;
            asm volatile("global_load_async_to_lds_b128 %0, %1, off"
                         :: "v"(lds_addr), "v"(gaddr)
                         : "memory");
        }
        asm volatile("s_wait_asynccnt 0x0" ::: "memory");
        __syncthreads();

        // Precompute gt areas once per tile (saves VALU in hot loop)
        for (int j = tid; j < cnt; j += BLOCK) {
            float4 g = s_gt[j];
            s_ag[j] = (g.z - g.x) * (g.w - g.y);
        }
        __syncthreads();

        // Hot loop: broadcast LDS reads; rcp (TRANS pipe) instead of IEEE div.
        #pragma unroll 4
        for (int j = 0; j < cnt; ++j) {
            float4 g  = s_gt[j];
            float ix0 = fmaxf(p.x, g.x);
            float iy0 = fmaxf(p.y, g.y);
            float ix1 = fminf(p.z, g.z);
            float iy1 = fminf(p.w, g.w);
            float iw  = fmaxf(ix1 - ix0, 0.0f);
            float ih  = fmaxf(iy1 - iy0, 0.0f);
            float inter = iw * ih;
            float uni   = area_p_eps + (s_ag[j] - inter);   // > 0 always (EPS)
            float iou   = inter * __builtin_amdgcn_rcpf(uni);
            max_iou = fmaxf(max_iou, iou);
        }
    }

    float loss = valid ? (1.0f - max_iou) : 0.0f;

    // ---- wave32 sum via the matrix unit ----
    // D = ones(16x4) * B + 0 with B-VGPR0 = loss, B-VGPR1 = 0: every lane of
    // D's first VGPR holds loss(n) + loss(n+16) for its column n. Four
    // xor-shuffles complete the 16-way sum within each lane half.
    v2f a; a[0] = 1.0f; a[1] = 1.0f;
    v2f b; b[0] = loss; b[1] = 0.0f;
    v8f c = {};
    v8f d = __builtin_amdgcn_wmma_f32_16x16x4_f32(
        /*neg_a=*/false, a, /*neg_b=*/false, b,
        /*c_mod=*/(short)0, c, /*reuse_a=*/false, /*reuse_b=*/false);
    float r = d[0];
    r += __shfl_xor(r, 1, 32);
    r += __shfl_xor(r, 2, 32);
    r += __shfl_xor(r, 4, 32);
    r += __shfl_xor(r, 8, 32);

    const int lane = tid & 31;
    const int wave = tid >> 5;
    if (lane == 0) s_wsum[wave] = r;
    __syncthreads();

    if (tid == 0) {
        float s = 0.0f;
        #pragma unroll
        for (int w = 0; w < BLOCK / 32; ++w) s += s_wsum[w];
        atomicAdd(out, s / (float)n_p);   // contributes block_sum / N -> mean
    }
}

extern "C" void kernel_launch(void* const* d_in, const int* in_sizes, int n_in,
                              void* d_out, int out_size, void* d_ws, size_t ws_size,
                              hipStream_t stream) {
    const float* proposals = (const float*)d_in[0];
    const float* gt_boxes  = (const float*)d_in[1];
    float* out = (float*)d_out;

    const int n_p = in_sizes[0] / 4;   // 50000 boxes (x4 coords)
    const int n_g = in_sizes[1] / 4;   // 1000 boxes

    // Deterministic: zero the accumulator every call (graph-capture safe).
    hipMemsetAsync(d_out, 0, (size_t)out_size * sizeof(float), stream);

    const int grid = (n_p + BLOCK - 1) / BLOCK;
    iou_loss_kernel<<<grid, BLOCK, 0, stream>>>(proposals, gt_boxes, out, n_p, n_g);
}